// Conv_12094627906068
// MI455X (gfx1250) — compile-verified
//
#include <hip/hip_runtime.h>
#include <stdint.h>

#define N_NODES  100000
#define N_EDGES  4000000
#define CHANNELS 64

typedef __attribute__((ext_vector_type(2))) float v2f;
typedef __attribute__((ext_vector_type(4))) float v4f;
typedef __attribute__((ext_vector_type(8))) float v8f;
typedef __attribute__((ext_vector_type(4))) unsigned int u32x4;
typedef __attribute__((ext_vector_type(8))) int i32x8;
typedef __attribute__((ext_vector_type(4))) int i32x4;

// ---------------------------------------------------------------------------
// Kernel 1: agg = x   (vectorized float4 copy into workspace accumulator)
// ---------------------------------------------------------------------------
__global__ void init_agg_kernel(const float* __restrict__ x,
                                float* __restrict__ agg) {
    int i = blockIdx.x * blockDim.x + threadIdx.x;   // one float4 per thread
    const int total = N_NODES * CHANNELS / 4;
    if (i < total) {
        ((v4f*)agg)[i] = ((const v4f*)x)[i];
    }
}

// ---------------------------------------------------------------------------
// Kernel 2: agg[targets[e]] += x[sources[e]]  (scatter-add)
// 16 lanes cooperate on one edge: each lane moves one float4 of the 256B row
// -> fully coalesced gather, then 4 native f32 atomic adds (relaxed, agent).
// x (25.6 MB) and agg (25.6 MB) both live in the 192 MB L2, so this phase
// runs at L2/atomic throughput, not HBM.
// ---------------------------------------------------------------------------
__global__ void scatter_add_kernel(const float* __restrict__ x,
                                   const int64_t* __restrict__ sources,
                                   const int64_t* __restrict__ targets,
                                   float* __restrict__ agg) {
    int gid = blockIdx.x * blockDim.x + threadIdx.x;
    int eid = gid >> 4;          // 16 lanes per edge
    int sub = gid & 15;          // which float4 of the 64-channel row
    if (eid >= N_EDGES) return;

    int64_t src = sources[eid];
    int64_t tgt = targets[eid];

    const float* srow = x + (size_t)src * CHANNELS + sub * 4;
    v4f v = *(const v4f*)srow;

    float* trow = agg + (size_t)tgt * CHANNELS + sub * 4;
    __hip_atomic_fetch_add(trow + 0, v.x, __ATOMIC_RELAXED, __HIP_MEMORY_SCOPE_AGENT);
    __hip_atomic_fetch_add(trow + 1, v.y, __ATOMIC_RELAXED, __HIP_MEMORY_SCOPE_AGENT);
    __hip_atomic_fetch_add(trow + 2, v.z, __ATOMIC_RELAXED, __HIP_MEMORY_SCOPE_AGENT);
    __hip_atomic_fetch_add(trow + 3, v.w, __ATOMIC_RELAXED, __HIP_MEMORY_SCOPE_AGENT);
}

// ---------------------------------------------------------------------------
// Kernel 3: out = (norm * agg) @ weight     via V_WMMA_F32_16X16X4_F32
//
// weight (64x64 f32 = 16 KB) is staged into LDS once per block by the Tensor
// Data Mover (TENSOR_LOAD_TO_LDS, 2D tile descriptor), then B fragments are
// served by ds_load -- keeping the VMEM pipe free for A loads and D stores.
//
// One wave32 computes a 16-row x 64-col output stripe:
//   A (16x4 f32): lanes 0-15 hold row M=lane, K=0..1; lanes 16-31 K=2..3.
//   B (4x16 f32): VGPR0 = {K=0 | K=2}, VGPR1 = {K=1 | K=3} striped over lanes.
//   C/D (16x16 f32, 8 VGPRs): VGPR v -> M = v + 8*(lane>=16), N = lane&15.
// K=64 -> 16 WMMA steps per N-tile, 4 N-tiles -> 64 v_wmma per wave.
// Compute guard is wave-uniform so EXEC is all-1s at every WMMA.
// ---------------------------------------------------------------------------
__global__ void gemm_norm_kernel(const float* __restrict__ agg,
                                 const float* __restrict__ norm,
                                 const float* __restrict__ weight,
                                 float* __restrict__ out) {
    __shared__ float wlds[CHANNELS * CHANNELS];          // 16 KB of 320 KB/WGP

    const int n_stripes = N_NODES / 16;                  // 6250 (exact)
    int lane   = threadIdx.x & 31;
    int stripe = blockIdx.x * (blockDim.x >> 5) + (threadIdx.x >> 5);

    // ---- TDM: DMA the whole 64x64 f32 weight tile global -> LDS (wave 0) ----
    if (threadIdx.x < 32) {
        uint64_t gaddr = (uint64_t)(uintptr_t)weight;
        uint32_t laddr = (uint32_t)(uintptr_t)&wlds[0];  // LDS offset = addr[31:0]

        u32x4 g0;
        g0.x = 1u;                                       // count=1, user D#, no gather
        g0.y = laddr;                                    // lds_addr
        g0.z = (uint32_t)gaddr;                          // global_addr[31:0]
        g0.w = ((uint32_t)(gaddr >> 32) & 0x01FFFFFFu)   // global_addr[56:32]
             | (2u << 30);                               // type=2 ("image")

        i32x8 g1;
        g1[0] = 0x00020000;                              // data_size=2 (4B), wg_mask=0
        g1[1] = (int)(64u << 16);                        // tensor_dim0 = 64
        g1[2] = (int)(64u << 16);                        // tensor_dim1 = 64
        g1[3] = (int)(64u << 16);                        // tile_dim0   = 64
        g1[4] = 64;                                      // tile_dim1   = 64
        g1[5] = 64;                                      // tensor_dim0_stride = 64
        g1[6] = 0;
        g1[7] = 0;

        i32x4 z4 = {0, 0, 0, 0};                         // groups 2/3: 2D tensor
        i32x8 z8 = {0, 0, 0, 0, 0, 0, 0, 0};             // 6-arg toolchain form
        __builtin_amdgcn_tensor_load_to_lds(g0, g1, z4, z4, z8, 0);
        __builtin_amdgcn_s_wait_tensorcnt(0);            // s_wait_tensorcnt 0
    }
    __syncthreads();                                     // publish tile to all waves

    if (stripe < n_stripes) {                            // wave-uniform guard
        int m0   = stripe * 16;
        int mrow = m0 + (lane & 15);
        int koff = (lane >> 4) * 2;                      // 0 for lanes 0-15, 2 for 16-31

        float scale = norm[mrow];

        // Full 16x64 A stripe: 16 k-steps, one v2f each, norm fused in.
        v2f a[16];
        const float* arow = agg + (size_t)mrow * CHANNELS;
        #pragma unroll
        for (int k = 0; k < 16; ++k) {
            v2f t = *(const v2f*)(arow + 4 * k + koff);
            a[k].x = t.x * scale;
            a[k].y = t.y * scale;
        }

        v8f c0 = {}, c1 = {}, c2 = {}, c3 = {};
        v8f* cacc[4] = { &c0, &c1, &c2, &c3 };

        #pragma unroll
        for (int nt = 0; nt < 4; ++nt) {
            int col = nt * 16 + (lane & 15);
            v8f acc = {};
            #pragma unroll
            for (int k = 0; k < 16; ++k) {
                int krow = 4 * k + koff;                 // K rows this lane supplies
                v2f b;
                b.x = wlds[krow * CHANNELS + col];       // ds_load (LDS-staged B)
                b.y = wlds[(krow + 1) * CHANNELS + col];
                // 8 args: (neg_a, A, neg_b, B, c_mod, C, reuse_a, reuse_b)
                acc = __builtin_amdgcn_wmma_f32_16x16x4_f32(
                    false, a[k], false, b, (short)0, acc, false, false);
            }
            *cacc[nt] = acc;
        }

        // Store D: VGPR v holds M = m0 + v + 8*(lane>=16), N = nt*16 + (lane&15)
        int rbase = m0 + (lane >> 4) * 8;
        int cbase = lane & 15;
        #pragma unroll
        for (int nt = 0; nt < 4; ++nt) {
            v8f acc = *cacc[nt];
            #pragma unroll
            for (int v = 0; v < 8; ++v) {
                out[(size_t)(rbase + v) * CHANNELS + nt * 16 + cbase] = acc[v];
            }
        }
    }
}

// ---------------------------------------------------------------------------
extern "C" void kernel_launch(void* const* d_in, const int* in_sizes, int n_in,
                              void* d_out, int out_size, void* d_ws, size_t ws_size,
                              hipStream_t stream) {
    const float*   x       = (const float*)d_in[0];     // [100000, 64]
    const int64_t* sources = (const int64_t*)d_in[1];   // [4M]
    const int64_t* targets = (const int64_t*)d_in[2];   // [4M]
    const float*   norm    = (const float*)d_in[3];     // [100000, 1]
    const float*   weight  = (const float*)d_in[4];     // [64, 64]
    float*         out     = (float*)d_out;             // [100000, 64]
    float*         agg     = (float*)d_ws;              // 25.6 MB accumulator

    (void)in_sizes; (void)n_in; (void)out_size; (void)ws_size;

    // 1) agg = x
    {
        int total  = N_NODES * CHANNELS / 4;            // 1.6M float4s
        int blocks = (total + 255) / 256;
        init_agg_kernel<<<blocks, 256, 0, stream>>>(x, agg);
    }

    // 2) agg[targets] += x[sources]
    {
        long long total = (long long)N_EDGES * 16;      // 64M threads
        int blocks = (int)((total + 255) / 256);        // 250000 blocks
        scatter_add_kernel<<<blocks, 256, 0, stream>>>(x, sources, targets, agg);
    }

    // 3) out = (norm * agg) @ weight   (TDM-staged B + WMMA fp32)
    {
        int n_stripes       = N_NODES / 16;             // 6250 stripes
        int waves_per_block = 8;                        // 256 threads
        int blocks = (n_stripes + waves_per_block - 1) / waves_per_block; // 782
        gemm_norm_kernel<<<blocks, 256, 0, stream>>>(agg, norm, weight, out);
    }
}